// AtomWise_23313082483613
// MI455X (gfx1250) — compile-verified
//
#include <hip/hip_runtime.h>
#include <hip/hip_bf16.h>

// Problem constants (match reference)
#define E_EDGES   300000
#define N_ATOMS   10000
#define C_CH      256
#define RBF_DIM   16

typedef __attribute__((ext_vector_type(2))) float v2f;
typedef __attribute__((ext_vector_type(8))) float v8f;

// D = A(16x4 f32) * B(4x16 f32) + C(16x16 f32), full-precision fp32 WMMA.
__device__ __forceinline__ v8f wmma_f32_16x16x4(v2f a, v2f b, v8f c) {
    return __builtin_amdgcn_wmma_f32_16x16x4_f32(
        /*neg_a=*/false, a, /*neg_b=*/false, b,
        /*c_mod=*/(short)0, c, /*reuse_a=*/false, /*reuse_b=*/false);
}

__device__ __forceinline__ float silu(float v) {
    return v / (1.0f + __expf(-v));
}

// ---------------------------------------------------------------------------
// Kernel 0: zero per-atom accumulator, seed output with b3.
// ---------------------------------------------------------------------------
__global__ __launch_bounds__(256) void init_kernel(float* __restrict__ acc,
                                                   float* __restrict__ out,
                                                   const float* __restrict__ b3) {
    const int i = blockIdx.x * 256 + threadIdx.x;
    if (i < N_ATOMS * C_CH) acc[i] = 0.0f;
    if (i < N_ATOMS)        out[i] = b3[0];
}

// ---------------------------------------------------------------------------
// Kernel 1 (streaming, HBM-bound): per 16-edge tile, one wave computes
// filter[16x256] = rbf[16x16] @ w_rbf^T (fp32 WMMA, bias in C), multiplies by
// x (non-temporal, batched loads), scatter-adds via global_atomic_add_f32.
// Base pointers hoisted so per-tile channel offsets fold into imm offsets.
// ---------------------------------------------------------------------------
__global__ __launch_bounds__(256) void edge_kernel(
    const float*     __restrict__ x,       // [E, C]
    const float*     __restrict__ rbf,     // [E, RBF]
    const long long* __restrict__ eidx,    // [E] int64
    const float*     __restrict__ w_rbf,   // [C, RBF]
    const float*     __restrict__ b_rbf,   // [C]
    float*           __restrict__ acc)     // [N_ATOMS, C]
{
    const int lane = threadIdx.x & 31;
    const int wave = threadIdx.x >> 5;
    const int tile = blockIdx.x * 8 + wave;
    constexpr int NT = E_EDGES / 16;       // 18750 tiles
    if (tile >= NT) return;                // wave-uniform: EXEC stays full

    const int e0   = tile * 16;
    const int half = lane >> 4;            // 0: lanes 0-15, 1: lanes 16-31
    const int row  = lane & 15;

    // A fragments: rbf rows e0..e0+15, K=0..15 in 4 steps of 4.
    v2f afrag[4];
#pragma unroll
    for (int ks = 0; ks < 4; ++ks) {
        const float* p = rbf + (size_t)(e0 + row) * RBF_DIM + ks * 4 + 2 * half;
        afrag[ks].x = p[0];
        afrag[ks].y = p[1];
    }

    // Hoisted per-lane base pointers for the 8 C-matrix rows this lane owns.
    const float* xbase[8];
    float*       abase[8];
#pragma unroll
    for (int v = 0; v < 8; ++v) {
        const int e = e0 + v + 8 * half;
        xbase[v] = x   + (size_t)e * C_CH + row;
        abase[v] = acc + (size_t)((int)eidx[e]) * C_CH + row;
    }

    const float* wbase = w_rbf + (size_t)row * RBF_DIM + 2 * half;
    const float* bbase = b_rbf + row;

#pragma unroll 4
    for (int nt = 0; nt < 16; ++nt) {
        const int n0 = nt * 16;

        // C init = bias (depends on N only -> same across the 8 row-VGPRs).
        const float bias = bbase[n0];
        v8f c;
#pragma unroll
        for (int v = 0; v < 8; ++v) c[v] = bias;

        // B fragments from w_rbf^T: B[k][n] = w_rbf[n][k].
#pragma unroll
        for (int ks = 0; ks < 4; ++ks) {
            const float* p = wbase + (size_t)n0 * RBF_DIM + ks * 4;
            v2f b;
            b.x = p[0];
            b.y = p[1];
            c = wmma_f32_16x16x4(afrag[ks], b, c);
        }

        // Batched gather of x (one clause + one wait), then 8 atomics.
        float xv[8];
#pragma unroll
        for (int v = 0; v < 8; ++v)
            xv[v] = __builtin_nontemporal_load(xbase[v] + n0);
#pragma unroll
        for (int v = 0; v < 8; ++v)
            unsafeAtomicAdd(abase[v] + n0, c[v] * xv[v]);  // global_atomic_add_f32
    }
}

// ---------------------------------------------------------------------------
// Kernel 2: h1 = silu(acc @ w1^T + b1).  One wave = 16 rows x 128 channels
// (8 N-tiles, 64 accumulator VGPRs); A-fragment reused across all 8 N-tiles.
// ---------------------------------------------------------------------------
__global__ __launch_bounds__(256) void mlp_silu_kernel(
    const float* __restrict__ in,    // [N_ATOMS, C]
    const float* __restrict__ w,     // [C, C]
    const float* __restrict__ bias,  // [C]
    float*       __restrict__ outbuf)// [N_ATOMS, C]
{
    const int lane = threadIdx.x & 31;
    const int wave = threadIdx.x >> 5;
    const int widx = blockIdx.x * 8 + wave;
    constexpr int NW = (N_ATOMS / 16) * 2;   // 1250 waves
    if (widx >= NW) return;                  // wave-uniform

    const int t    = widx >> 1;
    const int h    = widx & 1;               // which 128-channel half
    const int m0   = t * 16;
    const int half = lane >> 4;
    const int row  = lane & 15;

    v8f accv[8];
#pragma unroll
    for (int j = 0; j < 8; ++j) {
        const float b = bias[(h * 8 + j) * 16 + row];
#pragma unroll
        for (int v = 0; v < 8; ++v) accv[j][v] = b;
    }

    // Hoisted row pointers: A row and the 8 weight rows for our N-tiles.
    const float* arowp = in + (size_t)(m0 + row) * C_CH + 2 * half;
    const float* wrowp[8];
#pragma unroll
    for (int j = 0; j < 8; ++j)
        wrowp[j] = w + (size_t)((h * 8 + j) * 16 + row) * C_CH + 2 * half;

#pragma unroll 2
    for (int k = 0; k < C_CH; k += 4) {
        v2f a;
        a.x = arowp[k + 0];
        a.y = arowp[k + 1];
#pragma unroll
        for (int j = 0; j < 8; ++j) {
            v2f b;
            b.x = wrowp[j][k + 0];
            b.y = wrowp[j][k + 1];
            accv[j] = wmma_f32_16x16x4(a, b, accv[j]);
        }
    }

#pragma unroll
    for (int j = 0; j < 8; ++j) {
        const int n0 = (h * 8 + j) * 16;
#pragma unroll
        for (int v = 0; v < 8; ++v) {
            const int m = m0 + v + 8 * half;
            outbuf[(size_t)m * C_CH + n0 + row] = silu(accv[j][v]);
        }
    }
}

// ---------------------------------------------------------------------------
// Kernel 3: h2 = silu(h1 @ w2^T + b2) fused with out = h2 @ w3^T (+b3 seeded
// in init).  Shfl-xor reduction across the 16 lanes of each half-wave, then
// one float atomic per row (two waves contribute per row: the two C-halves).
// ---------------------------------------------------------------------------
__global__ __launch_bounds__(256) void mlp_head_kernel(
    const float* __restrict__ in,    // [N_ATOMS, C]  (h1)
    const float* __restrict__ w,     // [C, C]        (w2)
    const float* __restrict__ bias,  // [C]           (b2)
    const float* __restrict__ w3,    // [1, C]
    float*       __restrict__ out)   // [N_ATOMS]
{
    const int lane = threadIdx.x & 31;
    const int wave = threadIdx.x >> 5;
    const int widx = blockIdx.x * 8 + wave;
    constexpr int NW = (N_ATOMS / 16) * 2;
    if (widx >= NW) return;

    const int t    = widx >> 1;
    const int h    = widx & 1;
    const int m0   = t * 16;
    const int half = lane >> 4;
    const int row  = lane & 15;

    v8f accv[8];
#pragma unroll
    for (int j = 0; j < 8; ++j) {
        const float b = bias[(h * 8 + j) * 16 + row];
#pragma unroll
        for (int v = 0; v < 8; ++v) accv[j][v] = b;
    }

    const float* arowp = in + (size_t)(m0 + row) * C_CH + 2 * half;
    const float* wrowp[8];
#pragma unroll
    for (int j = 0; j < 8; ++j)
        wrowp[j] = w + (size_t)((h * 8 + j) * 16 + row) * C_CH + 2 * half;

#pragma unroll 2
    for (int k = 0; k < C_CH; k += 4) {
        v2f a;
        a.x = arowp[k + 0];
        a.y = arowp[k + 1];
#pragma unroll
        for (int j = 0; j < 8; ++j) {
            v2f b;
            b.x = wrowp[j][k + 0];
            b.y = wrowp[j][k + 1];
            accv[j] = wmma_f32_16x16x4(a, b, accv[j]);
        }
    }

    // Fused head: partial[m] = sum_n silu(h2[m][n]) * w3[n] over this wave's
    // 128 channels.
    float p[8];
#pragma unroll
    for (int v = 0; v < 8; ++v) p[v] = 0.0f;
#pragma unroll
    for (int j = 0; j < 8; ++j) {
        const float w3v = w3[(h * 8 + j) * 16 + row];
#pragma unroll
        for (int v = 0; v < 8; ++v)
            p[v] += silu(accv[j][v]) * w3v;
    }
    // Reduce over the 16 lanes of each half-wave (N dimension).
#pragma unroll
    for (int m = 8; m >= 1; m >>= 1) {
#pragma unroll
        for (int v = 0; v < 8; ++v) p[v] += __shfl_xor(p[v], m, 16);
    }
    if (row == 0) {
#pragma unroll
        for (int v = 0; v < 8; ++v)
            unsafeAtomicAdd(&out[m0 + v + 8 * half], p[v]);
    }
}

// ---------------------------------------------------------------------------
extern "C" void kernel_launch(void* const* d_in, const int* in_sizes, int n_in,
                              void* d_out, int out_size, void* d_ws, size_t ws_size,
                              hipStream_t stream) {
    (void)in_sizes; (void)n_in; (void)out_size; (void)ws_size;

    const float*     x     = (const float*)d_in[0];
    const float*     rbf   = (const float*)d_in[1];
    // d_in[2] = num_atoms (compile-time constant here)
    const long long* eidx  = (const long long*)d_in[3];
    const float*     w_rbf = (const float*)d_in[4];
    const float*     b_rbf = (const float*)d_in[5];
    const float*     w1    = (const float*)d_in[6];
    const float*     b1    = (const float*)d_in[7];
    const float*     w2    = (const float*)d_in[8];
    const float*     b2    = (const float*)d_in[9];
    const float*     w3    = (const float*)d_in[10];
    const float*     b3    = (const float*)d_in[11];
    float*           out   = (float*)d_out;

    float* acc = (float*)d_ws;                       // [N_ATOMS, C]
    float* h1  = acc + (size_t)N_ATOMS * C_CH;       // [N_ATOMS, C]

    {
        const int total  = N_ATOMS * C_CH;
        const int blocks = (total + 255) / 256;
        init_kernel<<<blocks, 256, 0, stream>>>(acc, out, b3);
    }
    {
        const int tiles  = E_EDGES / 16;             // 18750
        const int blocks = (tiles + 7) / 8;          // 8 waves / block
        edge_kernel<<<blocks, 256, 0, stream>>>(x, rbf, eidx, w_rbf, b_rbf, acc);
    }
    {
        const int waves  = (N_ATOMS / 16) * 2;       // 1250
        const int blocks = (waves + 7) / 8;
        mlp_silu_kernel<<<blocks, 256, 0, stream>>>(acc, w1, b1, h1);
        mlp_head_kernel<<<blocks, 256, 0, stream>>>(h1, w2, b2, w3, out);
    }
}